// Model_11665131176065
// MI455X (gfx1250) — compile-verified
//
#include <hip/hip_runtime.h>

// Problem constants (match reference)
#define NV  100000      // vertices
#define TT  128         // temporal depth
#define EE  1600000     // edges
#define KK  9           // conv kernel
#define NT  (NV * TT)   // 12,800,000

typedef __attribute__((ext_vector_type(2))) float v2f;
typedef __attribute__((ext_vector_type(8))) float v8f;

// ---------------------------------------------------------------- degree ----
__global__ void init_cnt_kernel(float* __restrict__ cnt) {
    int i = blockIdx.x * blockDim.x + threadIdx.x;
    if (i < NV) cnt[i] = 1.0f;              // self loop contributes 1
}

__global__ void degree_kernel(const int* __restrict__ dst, float* __restrict__ cnt) {
    int e = blockIdx.x * blockDim.x + threadIdx.x;
    if (e < EE) unsafeAtomicAdd(&cnt[dst[e]], 1.0f);
}

// ------------------------------------------------------------------ conv ----
// One block per node; 128 threads = one temporal row. Writes h and also
// initializes s = h (this is exactly the self-loop term of the scatter-sum).
__global__ __launch_bounds__(TT) void conv1d_kernel(
    const float* __restrict__ xin, const float* __restrict__ w,
    const float* __restrict__ b, float* __restrict__ h, float* __restrict__ s) {
    __shared__ float sx[TT];
    const int n = blockIdx.x;
    const int t = threadIdx.x;
    sx[t] = xin[(size_t)n * TT + t];
    __syncthreads();
    float acc = b[0];
#pragma unroll
    for (int k = 0; k < KK; ++k) {
        int tt = t + k - 4;                 // 'same' padding, PAD = 4
        float xv = (tt >= 0 && tt < TT) ? sx[tt] : 0.0f;
        acc = fmaf(xv, w[k], acc);
    }
    h[(size_t)n * TT + t] = acc;
    s[(size_t)n * TT + t] = acc;
}

// --------------------------------------------------------------- scatter ----
// One wave (32 lanes) covers one edge: lane handles 4 consecutive t values
// (float4 coalesced gather, 4x global_atomic_add_f32 scatter). src/dst are
// wave-uniform; 51.2MB h/s buffers are L2-resident (192MB L2).
__global__ __launch_bounds__(256) void scatter_kernel(
    const float* __restrict__ h, const int* __restrict__ src,
    const int* __restrict__ dst, float* __restrict__ s) {
    long long idx = (long long)blockIdx.x * 256 + threadIdx.x;
    if (idx >= (long long)EE * 32) return;
    const int e  = (int)(idx >> 5);
    const int t  = ((int)idx & 31) << 2;
    const int sn = src[e];
    const int dn = dst[e];
    const float4 v = *(const float4*)(h + (size_t)sn * TT + t);
    float* p = s + (size_t)dn * TT + t;
    unsafeAtomicAdd(p + 0, v.x);
    unsafeAtomicAdd(p + 1, v.y);
    unsafeAtomicAdd(p + 2, v.z);
    unsafeAtomicAdd(p + 3, v.w);
}

// ------------------------------------------------------------- mean+relu ----
__global__ void relu_mean_kernel(const float* __restrict__ s,
                                 const float* __restrict__ cnt,
                                 float* __restrict__ x) {
    int i = blockIdx.x * blockDim.x + threadIdx.x;   // i < NT (exact grid)
    int n = i >> 7;                                   // / TT
    x[i] = fmaxf(s[i] / cnt[n], 0.0f);
}

// -------------------------------------------------------------- out init ----
__global__ void init_out_kernel(const float* __restrict__ bout, float* __restrict__ out) {
    int i = blockIdx.x * blockDim.x + threadIdx.x;
    if (i < TT * 3) out[i] = bout[i % 3];
}

// ------------------------------------------------------------ WMMA gemm -----
// out[t,c] += sum_n xflat[t*NV + n] * W[c*NV + n]
// A = 16x4 f32 tile (M=t rows), B = 4x16 (cols = c; only c<3 stored, W row
// pointer clamped so unstored columns just compute harmless values).
// Slot->k remap: per 8-k pair, lane-half0 supplies slots{0,1} from k..k+3 and
// half1 supplies slots{2,3} from k+4..k+7, so ONE b128 load feeds TWO WMMAs.
// 16 k per iteration = 4 b128 loads + 4 independent-accumulator WMMAs.
#define KSLICE 2000   // 50 slices; 2000 % 16 == 0 -> branch-free inner loop
__global__ __launch_bounds__(32) void out_gemm_wmma_kernel(
    const float* __restrict__ xf, const float* __restrict__ W,
    float* __restrict__ out) {
    const int lane = threadIdx.x;
    const int tile = blockIdx.x;            // 8 tiles of 16 t-rows
    const int half = lane >> 4;             // 0 or 1 (lane group)
    const int row  = tile * 16 + (lane & 15);
    const int c    = lane & 15;
    const float* ap = xf + (size_t)row * NV + half * 4;
    const float* wp = W + (size_t)((c < 3) ? c : 2) * NV + half * 4;

    v8f acc0 = {}, acc1 = {}, acc2 = {}, acc3 = {};
    const int k0 = blockIdx.y * KSLICE;
    const int k1 = k0 + KSLICE;             // 50 * 2000 == NV exactly
    for (int k = k0; k < k1; k += 16) {
        const float4 a0 = *(const float4*)(ap + k);
        const float4 b0 = *(const float4*)(wp + k);
        const float4 a1 = *(const float4*)(ap + k + 8);
        const float4 b1 = *(const float4*)(wp + k + 8);
        v2f aA = {a0.x, a0.y}, bA = {b0.x, b0.y};
        v2f aB = {a0.z, a0.w}, bB = {b0.z, b0.w};
        v2f aC = {a1.x, a1.y}, bC = {b1.x, b1.y};
        v2f aD = {a1.z, a1.w}, bD = {b1.z, b1.w};
        acc0 = __builtin_amdgcn_wmma_f32_16x16x4_f32(false, aA, false, bA, (short)0, acc0, false, false);
        acc1 = __builtin_amdgcn_wmma_f32_16x16x4_f32(false, aB, false, bB, (short)0, acc1, false, false);
        acc2 = __builtin_amdgcn_wmma_f32_16x16x4_f32(false, aC, false, bC, (short)0, acc2, false, false);
        acc3 = __builtin_amdgcn_wmma_f32_16x16x4_f32(false, aD, false, bD, (short)0, acc3, false, false);
    }
    v8f tot = (acc0 + acc1) + (acc2 + acc3);
    if (c < 3) {
#pragma unroll
        for (int v = 0; v < 8; ++v) {
            int tl = half * 8 + v;          // D: VGPR v holds M = v (+8 hi half)
            unsafeAtomicAdd(&out[(size_t)(tile * 16 + tl) * 3 + c], tot[v]);
        }
    }
}

// ----------------------------------------------------------------- driver ---
extern "C" void kernel_launch(void* const* d_in, const int* in_sizes, int n_in,
                              void* d_out, int out_size, void* d_ws, size_t ws_size,
                              hipStream_t stream) {
    const float* x    = (const float*)d_in[0];
    const int*   ei   = (const int*)  d_in[1];
    const float* cw   = (const float*)d_in[2];   // [3,1,1,9]
    const float* cb   = (const float*)d_in[3];   // [3,1]
    const float* Wout = (const float*)d_in[4];   // [3, NV]
    const float* bout = (const float*)d_in[5];   // [3]
    float* out = (float*)d_out;                  // [TT, 3]

    const int* src = ei;                         // edge_index[0]
    const int* dst = ei + EE;                    // edge_index[1]

    float* ws  = (float*)d_ws;
    float* cnt = ws;                             // NV floats
    float* hb  = ws + (1 << 17);                 // NT floats
    float* sb  = hb + NT;                        // NT floats
    float* xb  = sb + NT;                        // NT floats

    init_cnt_kernel<<<(NV + 255) / 256, 256, 0, stream>>>(cnt);
    degree_kernel<<<(EE + 255) / 256, 256, 0, stream>>>(dst, cnt);

    for (int i = 0; i < 3; ++i) {
        const float* xin = (i == 0) ? x : xb;
        conv1d_kernel<<<NV, TT, 0, stream>>>(xin, cw + i * KK, cb + i, hb, sb);
        scatter_kernel<<<(int)(((long long)EE * 32 + 255) / 256), 256, 0, stream>>>(
            hb, src, dst, sb);
        relu_mean_kernel<<<NT / 256, 256, 0, stream>>>(sb, cnt, xb);
    }

    init_out_kernel<<<2, 256, 0, stream>>>(bout, out);
    out_gemm_wmma_kernel<<<dim3(8, NV / KSLICE), 32, 0, stream>>>(xb, Wout, out);
}